// CompletionNet_11209864643009
// MI455X (gfx1250) — compile-verified
//
#include <hip/hip_runtime.h>
#include <math.h>

#define BN_EPS 1e-5f

typedef __attribute__((ext_vector_type(2))) float v2f;
typedef __attribute__((ext_vector_type(8))) float v8f;

// ---------------------------------------------------------------------------
// Implicit-GEMM 3x3x3 stride-1 SAME conv + BN + ReLU (+ optional skip add).
// One wave computes TWO 16-voxel x 16-cout tiles (M=32) with
// V_WMMA_F32_16X16X4_F32, sharing the B (weight) fragment between them.
// CIN/COUT are compile-time so the K-chunk loop fully unrolls and all
// channel predicates fold to constants. All global loads are unconditional
// (clamped-safe address), only the *values* are masked with v_cndmask:
//  - A values zeroed for spatial OOB; channel-pad K rows are neutralized by
//    constant-zero B rows.
//  - Columns ncol>=COUT carry garbage through the WMMA but are never stored.
// total output rows M = 2*D^3: 16 at D=2 (tile1 clamped), else multiple of 32.
// ---------------------------------------------------------------------------
template <int CIN, int COUT>
__global__ __launch_bounds__(32)
void conv3_wmma(const float* __restrict__ in,
                const float* __restrict__ w,      // (COUT,CIN,3,3,3)
                const float* __restrict__ gamma,
                const float* __restrict__ beta,
                const float* __restrict__ mean,
                const float* __restrict__ var,
                const float* __restrict__ skip,   // may be null
                float* __restrict__ out,
                int ld /* log2(D) */)
{
    const int lane = threadIdx.x;
    const int D    = 1 << ld;
    const int vol  = 1 << (3 * ld);
    const int total_m = 2 * vol;
    const int hi   = lane >> 4;        // lane half selects K pair {0,1} vs {2,3}
    const int kb   = hi * 2;
    const int ncol = lane & 15;        // B/C/D column = cout index

    // Two A-tile rows per wave; tile1 clamped to tile0 when out of range.
    const int  m0  = blockIdx.x * 32 + ncol;
    const bool t1v = blockIdx.x * 32 + 16 < total_m;
    const int  m1  = t1v ? m0 + 16 : m0;

    const int n0 = m0 >> (3 * ld), v0 = m0 & (vol - 1);
    const int z0 = v0 >> (2 * ld), y0 = (v0 >> ld) & (D - 1), x0 = v0 & (D - 1);
    const int n1 = m1 >> (3 * ld), v1 = m1 & (vol - 1);
    const int z1 = v1 >> (2 * ld), y1 = (v1 >> ld) & (D - 1), x1 = v1 & (D - 1);

    // Hoisted per-axis bounds (tap coordinate = c + t - 1, t in 0..2)
    bool bz0[3], by0[3], bx0[3], bz1[3], by1[3], bx1[3];
    #pragma unroll
    for (int t = 0; t < 3; ++t) {
        bz0[t] = (unsigned)(z0 + t - 1) < (unsigned)D;
        by0[t] = (unsigned)(y0 + t - 1) < (unsigned)D;
        bx0[t] = (unsigned)(x0 + t - 1) < (unsigned)D;
        bz1[t] = (unsigned)(z1 + t - 1) < (unsigned)D;
        by1[t] = (unsigned)(y1 + t - 1) < (unsigned)D;
        bx1[t] = (unsigned)(x1 + t - 1) < (unsigned)D;
    }

    const bool hw = ncol < COUT;
    const int  wn = hw ? ncol : 0;

    v8f acc0 = {}, acc1 = {};
    #pragma unroll
    for (int cb = 0; cb < CIN; cb += 4) {
        const int  c0  = cb + kb;                 // kb in {0,2} per lane half
        const int  c1  = c0 + 1;
        const bool hc0 = c0 < CIN;                // kb-dependent, lane-uniform
        const bool hc1 = c1 < CIN;
        const int  sc0 = hc0 ? c0 : 0;            // clamped-safe channels
        const int  sc1 = hc1 ? c1 : 0;

        const int ba00 = ((n0 * CIN + sc0) << (3 * ld)) + v0;
        const int ba01 = ((n0 * CIN + sc1) << (3 * ld)) + v0;
        const int ba10 = ((n1 * CIN + sc0) << (3 * ld)) + v1;
        const int ba11 = ((n1 * CIN + sc1) << (3 * ld)) + v1;
        const int wb0  = (wn * CIN + sc0) * 27;
        const int wb1  = (wn * CIN + sc1) * 27;

        #pragma unroll
        for (int tz = 0; tz < 3; ++tz)
        #pragma unroll
        for (int ty = 0; ty < 3; ++ty)
        #pragma unroll
        for (int tx = 0; tx < 3; ++tx) {
            const int t   = (tz * 3 + ty) * 3 + tx;
            const int off = ((tz - 1) * D + (ty - 1)) * D + (tx - 1);

            const bool ib0 = bz0[tz] && by0[ty] && bx0[tx];
            const bool ib1 = bz1[tz] && by1[ty] && bx1[tx];
            const int  o0  = ib0 ? off : -v0;   // fallback -> channel base
            const int  o1  = ib1 ? off : -v1;

            // Unconditional loads from safe addresses (never predicated).
            const float ta00 = in[ba00 + o0];
            const float ta01 = in[ba01 + o0];
            const float ta10 = in[ba10 + o1];
            const float ta11 = in[ba11 + o1];
            const float tw0  = hc0 ? w[wb0 + t] : 0.f;  // constexpr predicate
            const float tw1  = hc1 ? w[wb1 + t] : 0.f;

            v2f a0, a1, b;
            a0.x = ib0 ? ta00 : 0.f;
            a0.y = ib0 ? ta01 : 0.f;
            a1.x = ib1 ? ta10 : 0.f;
            a1.y = ib1 ? ta11 : 0.f;
            b.x  = tw0;
            b.y  = tw1;

            acc0 = __builtin_amdgcn_wmma_f32_16x16x4_f32(
                       false, a0, false, b, (short)0, acc0, false, false);
            acc1 = __builtin_amdgcn_wmma_f32_16x16x4_f32(
                       false, a1, false, b, (short)0, acc1, false, false);
        }
    }

    // Epilogue: BN + ReLU (+skip). Lane holds cout=ncol over 8 M rows per tile.
    if (hw) {
        const float scv = gamma[ncol] * rsqrtf(var[ncol] + BN_EPS);
        const float shv = beta[ncol] - mean[ncol] * scv;
        #pragma unroll
        for (int j = 0; j < 8; ++j) {
            {
                const int mr = blockIdx.x * 32 + j + hi * 8;
                const int nn = mr >> (3 * ld), vv = mr & (vol - 1);
                float val = fmaxf(acc0[j] * scv + shv, 0.f);
                const int oi = ((nn * COUT + ncol) << (3 * ld)) + vv;
                if (skip) val += skip[oi];
                out[oi] = val;
            }
            if (t1v) {
                const int mr = blockIdx.x * 32 + 16 + j + hi * 8;
                const int nn = mr >> (3 * ld), vv = mr & (vol - 1);
                float val = fmaxf(acc1[j] * scv + shv, 0.f);
                const int oi = ((nn * COUT + ncol) << (3 * ld)) + vv;
                if (skip) val += skip[oi];
                out[oi] = val;
            }
        }
    }
}

// ---------------------------------------------------------------------------
// 2x2x2 stride-2 VALID conv + BN + ReLU (encoder downsample), scalar.
// ---------------------------------------------------------------------------
__global__ void down2_bn_relu(const float* __restrict__ in,
                              const float* __restrict__ w,   // (Cout,Cin,2,2,2)
                              const float* __restrict__ gamma,
                              const float* __restrict__ beta,
                              const float* __restrict__ mean,
                              const float* __restrict__ var,
                              float* __restrict__ out,
                              int Cin, int Cout, int Din, int total)
{
    int i = blockIdx.x * blockDim.x + threadIdx.x;
    if (i >= total) return;
    const int Do = Din >> 1;
    int x = i % Do; int r = i / Do;
    int y = r % Do; r /= Do;
    int z = r % Do; r /= Do;
    int co = r % Cout; int nn = r / Cout;

    float s = 0.f;
    for (int ci = 0; ci < Cin; ++ci) {
        const float* wp = w + (co * Cin + ci) * 8;
        const long base = ((long)(nn * Cin + ci)) * Din * Din * Din;
        #pragma unroll
        for (int tz = 0; tz < 2; ++tz)
        #pragma unroll
        for (int ty = 0; ty < 2; ++ty)
        #pragma unroll
        for (int tx = 0; tx < 2; ++tx)
            s += wp[(tz * 2 + ty) * 2 + tx] *
                 in[base + ((long)(2 * z + tz) * Din + (2 * y + ty)) * Din + (2 * x + tx)];
    }
    const float sc = gamma[co] * rsqrtf(var[co] + BN_EPS);
    out[i] = fmaxf(s * sc + (beta[co] - mean[co] * sc), 0.f);
}

// ---------------------------------------------------------------------------
// Transposed conv stride 2 (jax.lax.conv_transpose, transpose_kernel=False:
// correlation on lhs-dilated input, pad = K-1-p per side) + BN + ReLU.
// K=4 with p=1 (P=2) or K=2 with p=0 (P=1).  Dout = 2*Din.
// ---------------------------------------------------------------------------
__global__ void up_bn_relu(const float* __restrict__ in,
                           const float* __restrict__ w,  // (Cout,Cin,K,K,K)
                           const float* __restrict__ gamma,
                           const float* __restrict__ beta,
                           const float* __restrict__ mean,
                           const float* __restrict__ var,
                           float* __restrict__ out,
                           int Cin, int Cout, int Din, int K, int total)
{
    int i = blockIdx.x * blockDim.x + threadIdx.x;
    if (i >= total) return;
    const int Do = Din * 2;
    int x = i % Do; int r = i / Do;
    int y = r % Do; r /= Do;
    int z = r % Do; r /= Do;
    int co = r % Cout; int nn = r / Cout;

    const int P    = (K == 4) ? 2 : 1;     // K-1-p
    const int vmax = 2 * Din - 2;
    float s = 0.f;
    for (int ci = 0; ci < Cin; ++ci) {
        const float* wp = w + ((co * Cin + ci) * K) * K * K;
        const long base = ((long)(nn * Cin + ci)) * Din * Din * Din;
        for (int tz = 0; tz < K; ++tz) {
            const int vz = z + tz - P;
            if (vz < 0 || vz > vmax || (vz & 1)) continue;
            for (int ty = 0; ty < K; ++ty) {
                const int vy = y + ty - P;
                if (vy < 0 || vy > vmax || (vy & 1)) continue;
                for (int tx = 0; tx < K; ++tx) {
                    const int vx = x + tx - P;
                    if (vx < 0 || vx > vmax || (vx & 1)) continue;
                    s += wp[(tz * K + ty) * K + tx] *
                         in[base + ((long)(vz >> 1) * Din + (vy >> 1)) * Din + (vx >> 1)];
                }
            }
        }
    }
    const float sc = gamma[co] * rsqrtf(var[co] + BN_EPS);
    out[i] = fmaxf(s * sc + (beta[co] - mean[co] * sc), 0.f);
}

// ---------------------------------------------------------------------------
// Fused 1x1x1 classifier + pruning mask, in place: d *= (cls(d) > 0).
// ---------------------------------------------------------------------------
__global__ void cls_mask(float* __restrict__ d,
                         const float* __restrict__ w,  // (1,C,1,1,1)
                         const float* __restrict__ b,  // (1,)
                         int C, int vol, int total)
{
    int i = blockIdx.x * blockDim.x + threadIdx.x;
    if (i >= total) return;
    const int nn = i / vol, vx = i % vol;
    float logit = b[0];
    for (int c = 0; c < C; ++c)
        logit += w[c] * d[((long)(nn * C + c)) * vol + vx];
    if (!(logit > 0.f))
        for (int c = 0; c < C; ++c)
            d[((long)(nn * C + c)) * vol + vx] = 0.f;
}

// ---------------------------------------------------------------------------
// Final: sigmoid(cls(d)) -> d_out
// ---------------------------------------------------------------------------
__global__ void final_sigmoid(const float* __restrict__ d,
                              const float* __restrict__ w,
                              const float* __restrict__ b,
                              float* __restrict__ out,
                              int C, int vol, int total)
{
    int i = blockIdx.x * blockDim.x + threadIdx.x;
    if (i >= total) return;
    const int nn = i / vol, vx = i % vol;
    float logit = b[0];
    for (int c = 0; c < C; ++c)
        logit += w[c] * d[((long)(nn * C + c)) * vol + vx];
    out[i] = 1.f / (1.f + expf(-logit));
}

// ---------------------------------------------------------------------------
// Host orchestration
// ---------------------------------------------------------------------------
static void launch_conv3(const float* in, const float* w,
                         const float* g, const float* be,
                         const float* me, const float* va,
                         const float* skip, float* out,
                         int Cin, int Cout, int ld, hipStream_t stream)
{
    const int total_m = 2 << (3 * ld);
    const int tiles = (total_m + 31) / 32;
    if (Cin == 1 && Cout == 2)
        conv3_wmma<1, 2><<<tiles, 32, 0, stream>>>(in, w, g, be, me, va, skip, out, ld);
    else if (Cin == 2 && Cout == 2)
        conv3_wmma<2, 2><<<tiles, 32, 0, stream>>>(in, w, g, be, me, va, skip, out, ld);
    else if (Cin == 4 && Cout == 4)
        conv3_wmma<4, 4><<<tiles, 32, 0, stream>>>(in, w, g, be, me, va, skip, out, ld);
    else
        conv3_wmma<16, 16><<<tiles, 32, 0, stream>>>(in, w, g, be, me, va, skip, out, ld);
}

extern "C" void kernel_launch(void* const* d_in, const int* in_sizes, int n_in,
                              void* d_out, int out_size, void* d_ws, size_t ws_size,
                              hipStream_t stream) {
    (void)in_sizes; (void)n_in; (void)out_size; (void)ws_size;
    const float* x = (const float*)d_in[0];
    auto P = [&](int i) { return (const float*)d_in[i]; };

    static const int CH[7] = {2, 2, 4, 4, 4, 4, 16};

    // workspace layout (floats)
    float* ws = (float*)d_ws;
    size_t off = 0;
    float* enc[7];
    const size_t encSize[7] = {
        (size_t)2 * 2 * 128 * 128 * 128,   // enc0
        (size_t)2 * 2 * 64 * 64 * 64,      // enc1
        (size_t)2 * 4 * 32 * 32 * 32,      // enc2
        (size_t)2 * 4 * 16 * 16 * 16,      // enc3
        (size_t)2 * 4 * 8 * 8 * 8,         // enc4
        (size_t)2 * 4 * 4 * 4 * 4,         // enc5
        (size_t)2 * 16 * 2 * 2 * 2         // enc6
    };
    for (int i = 0; i < 7; ++i) { enc[i] = ws + off; off += encSize[i]; }
    float* tmp0 = ws + off; off += (size_t)2 * 2 * 128 * 128 * 128;
    float* tmp1 = ws + off; off += (size_t)2 * 2 * 128 * 128 * 128;

    // ---- encoder ----
    int ld = 7;  // D = 128
    launch_conv3(x, P(1), P(2), P(3), P(4), P(5), nullptr, enc[0], 1, CH[0], ld, stream);
    for (int i = 0; i < 6; ++i) {
        const int base = 6 + i * 10;
        const int Cin = CH[i], Cout = CH[i + 1];
        const int Din = 1 << ld;
        const int ldo = ld - 1;
        const int Do  = 1 << ldo;
        const int totalD = 2 * Cout * Do * Do * Do;
        down2_bn_relu<<<(totalD + 255) / 256, 256, 0, stream>>>(
            enc[i], P(base + 0), P(base + 1), P(base + 2), P(base + 3), P(base + 4),
            tmp0, Cin, Cout, Din, totalD);
        launch_conv3(tmp0, P(base + 5), P(base + 6), P(base + 7), P(base + 8), P(base + 9),
                     nullptr, enc[i + 1], Cout, Cout, ldo, stream);
        ld = ldo;
    }

    // ---- decoder ----  (ld == 1, D = 2, d = enc[6])
    float* d = enc[6];
    for (int i = 0; i < 6; ++i) {
        const int base = 66 + i * 12;
        const int Cin = CH[6 - i], Cout = CH[5 - i];
        const int K = (i == 0) ? 4 : 2;
        const int Din = 1 << ld;
        const int ldo = ld + 1;
        const int Do  = 1 << ldo;
        const int totalU = 2 * Cout * Do * Do * Do;
        up_bn_relu<<<(totalU + 255) / 256, 256, 0, stream>>>(
            d, P(base + 0), P(base + 1), P(base + 2), P(base + 3), P(base + 4),
            tmp0, Cin, Cout, Din, K, totalU);
        launch_conv3(tmp0, P(base + 5), P(base + 6), P(base + 7), P(base + 8), P(base + 9),
                     enc[5 - i], tmp1, Cout, Cout, ldo, stream);
        const int vol = Do * Do * Do;
        const int totalV = 2 * vol;
        cls_mask<<<(totalV + 255) / 256, 256, 0, stream>>>(
            tmp1, P(base + 10), P(base + 11), Cout, vol, totalV);
        d = tmp1;
        ld = ldo;
    }

    // ---- final ----  (ld == 7, C = 2)
    {
        const int vol = 1 << 21;
        const int totalV = 2 * vol;
        final_sigmoid<<<(totalV + 255) / 256, 256, 0, stream>>>(
            d, P(138), P(139), (float*)d_out, CH[0], vol, totalV);
    }
}